// DepthNet_10350871184003
// MI455X (gfx1250) — compile-verified
//
#include <hip/hip_runtime.h>
#include <math.h>

typedef float v2f __attribute__((ext_vector_type(2)));
typedef float v4f __attribute__((ext_vector_type(4)));
typedef float v8f __attribute__((ext_vector_type(8)));

#define B_   2
#define C_   32
#define H_   96
#define W_   128
#define D_   48
#define NSRC 2
#define NV   (NSRC + 1)
#define HW_  (H_ * W_)

// ---------------------------------------------------------------------------
// Kernel 0: combine projections and build per-(view,batch) rot(3x3)+trans(3).
// combined = [K*R, K*t; 0 0 0 1];  proj = src_c * inv(ref_c)
// inv([A t;0 1]) = [A^-1, -A^-1 t; 0 1]  ->  rot = As*Ar^-1, trans = ts - rot*tr
// ---------------------------------------------------------------------------
__global__ void k_proj(const float* __restrict__ ref_proj,
                       const float* __restrict__ src_projs,
                       float* __restrict__ prm) {
  int t = threadIdx.x;
  if (t >= NSRC * B_) return;
  int b = t % B_;
  const float* rp = ref_proj + b * 32;  // [2][4][4]
  const float* sp = src_projs + t * 32; // flat (v*B+b)*[2][4][4]

  float Ar[9], tr[3], As[9], ts[3];
  for (int i = 0; i < 3; i++) {
    for (int j = 0; j < 3; j++) {
      float a = 0.f, s = 0.f;
      for (int k = 0; k < 3; k++) {
        a += rp[16 + i * 4 + k] * rp[k * 4 + j];
        s += sp[16 + i * 4 + k] * sp[k * 4 + j];
      }
      Ar[i * 3 + j] = a; As[i * 3 + j] = s;
    }
    float a = 0.f, s = 0.f;
    for (int k = 0; k < 3; k++) {
      a += rp[16 + i * 4 + k] * rp[k * 4 + 3];
      s += sp[16 + i * 4 + k] * sp[k * 4 + 3];
    }
    tr[i] = a; ts[i] = s;
  }
  // 3x3 inverse of Ar
  float a0 = Ar[0], a1 = Ar[1], a2 = Ar[2];
  float a3 = Ar[3], a4 = Ar[4], a5 = Ar[5];
  float a6 = Ar[6], a7 = Ar[7], a8 = Ar[8];
  float det = a0 * (a4 * a8 - a5 * a7) - a1 * (a3 * a8 - a5 * a6) + a2 * (a3 * a7 - a4 * a6);
  float id = 1.0f / det;
  float Ai[9];
  Ai[0] = (a4 * a8 - a5 * a7) * id;
  Ai[1] = (a2 * a7 - a1 * a8) * id;
  Ai[2] = (a1 * a5 - a2 * a4) * id;
  Ai[3] = (a5 * a6 - a3 * a8) * id;
  Ai[4] = (a0 * a8 - a2 * a6) * id;
  Ai[5] = (a2 * a3 - a0 * a5) * id;
  Ai[6] = (a3 * a7 - a4 * a6) * id;
  Ai[7] = (a1 * a6 - a0 * a7) * id;
  Ai[8] = (a0 * a4 - a1 * a3) * id;

  float* o = prm + t * 12;
  float rot[9];
  for (int i = 0; i < 3; i++)
    for (int j = 0; j < 3; j++) {
      float s = 0.f;
      for (int k = 0; k < 3; k++) s += As[i * 3 + k] * Ai[k * 3 + j];
      rot[i * 3 + j] = s;
      o[i * 3 + j] = s;
    }
  for (int i = 0; i < 3; i++)
    o[9 + i] = ts[i] - (rot[i * 3 + 0] * tr[0] + rot[i * 3 + 1] * tr[1] + rot[i * 3 + 2] * tr[2]);
}

// ---------------------------------------------------------------------------
// Kernel 1: fused warp + variance + channel-dot (via V_WMMA_F32_16X16X4_F32).
// One wave = 16 consecutive (w) points; lane l owns point m=l&15 and the 16
// channels c with bit1(c)==l>>4, matching the 16x4 f32 A-operand layout, so
// 8 chained WMMAs compute cost[m] = sum_c var[m,c]*w_reg[c] with zero
// cross-lane shuffles. D columns are all equal (B replicated) -> lanes 0/16
// store 8 contiguous floats each.
// ---------------------------------------------------------------------------
__global__ void __launch_bounds__(256) k_cost(
    const float* __restrict__ ref_fea,
    const float* __restrict__ src_fea,
    const float* __restrict__ depth_values,
    const float* __restrict__ w_reg,
    const float* __restrict__ b_reg,
    const float* __restrict__ prm,
    float* __restrict__ cost) {
  const int lane = threadIdx.x & 31;
  const int hi = lane >> 4;   // which half of the wave
  const int m = lane & 15;    // point within wave's 16-point tile
  const int wave = blockIdx.x * (blockDim.x >> 5) + (threadIdx.x >> 5);
  const int pbase = wave << 4;
  const int p = pbase + m;

  const int w = p & (W_ - 1);
  int r = p >> 7;                 // / W_
  const int h = r % H_; r /= H_;
  const int d = r % D_;
  const int b = r / D_;

  const float dep = depth_values[b * D_ + d];
  const float x = (float)w, y = (float)h;

  float tw[NSRC][4];
  int to[NSRC][4];
  int tb[NSRC];
#pragma unroll
  for (int v = 0; v < NSRC; ++v) {
    const float* pr = prm + (v * B_ + b) * 12;
    float rx = pr[0] * x + pr[1] * y + pr[2];
    float ry = pr[3] * x + pr[4] * y + pr[5];
    float rz = pr[6] * x + pr[7] * y + pr[8];
    float X = rx * dep + pr[9];
    float Y = ry * dep + pr[10];
    float Z = rz * dep + pr[11];
    float iz = 1.0f / Z;
    float px = X * iz, py = Y * iz;
    float x0f = floorf(px), y0f = floorf(py);
    float wx = px - x0f, wy = py - y0f;
    int x0 = (int)x0f, y0 = (int)y0f;
    int x1 = x0 + 1, y1 = y0 + 1;
    float vx0 = (x0 >= 0 && x0 < W_) ? 1.f : 0.f;
    float vx1 = (x1 >= 0 && x1 < W_) ? 1.f : 0.f;
    float vy0 = (y0 >= 0 && y0 < H_) ? 1.f : 0.f;
    float vy1 = (y1 >= 0 && y1 < H_) ? 1.f : 0.f;
    int x0c = min(max(x0, 0), W_ - 1), x1c = min(max(x1, 0), W_ - 1);
    int y0c = min(max(y0, 0), H_ - 1), y1c = min(max(y1, 0), H_ - 1);
    tw[v][0] = (1.f - wx) * (1.f - wy) * vx0 * vy0;
    tw[v][1] = wx * (1.f - wy) * vx1 * vy0;
    tw[v][2] = (1.f - wx) * wy * vx0 * vy1;
    tw[v][3] = wx * wy * vx1 * vy1;
    to[v][0] = y0c * W_ + x0c;
    to[v][1] = y0c * W_ + x1c;
    to[v][2] = y1c * W_ + x0c;
    to[v][3] = y1c * W_ + x1c;
    tb[v] = (v * B_ + b) * C_ * HW_;
  }

  const int refbase = b * C_ * HW_ + h * W_ + w;
  const float invV = 1.0f / (float)NV;

  v8f acc = {0.f, 0.f, 0.f, 0.f, 0.f, 0.f, 0.f, 0.f};
#pragma unroll
  for (int j = 0; j < 8; ++j) {
    v2f A, Bv;
#pragma unroll
    for (int t = 0; t < 2; ++t) {
      const int c = 4 * j + 2 * hi + t;   // channel mapped to A[m, k] layout
      float s = ref_fea[refbase + c * HW_];
      float q = s * s;
#pragma unroll
      for (int v = 0; v < NSRC; ++v) {
        const float* f = src_fea + tb[v] + c * HW_;
        float wv = tw[v][0] * f[to[v][0]] + tw[v][1] * f[to[v][1]] +
                   tw[v][2] * f[to[v][2]] + tw[v][3] * f[to[v][3]];
        s += wv;
        q += wv * wv;
      }
      float sV = s * invV;
      A[t] = q * invV - sV * sV;          // variance(m, c)
      Bv[t] = w_reg[c];                   // same channel, replicated over N
    }
    acc = __builtin_amdgcn_wmma_f32_16x16x4_f32(
        false, A, false, Bv, (short)0, acc, false, false);
  }

  const float bb = b_reg[0];
  if (m == 0) {  // lanes 0 (rows 0..7) and 16 (rows 8..15) hold the results
    float* dst = cost + pbase + 8 * hi;
    v4f lo = {acc[0] + bb, acc[1] + bb, acc[2] + bb, acc[3] + bb};
    v4f hi4 = {acc[4] + bb, acc[5] + bb, acc[6] + bb, acc[7] + bb};
    *(v4f*)(dst) = lo;
    *(v4f*)(dst + 4) = hi4;
  }
}

// ---------------------------------------------------------------------------
// Kernel 2: per-pixel softmax over D, expected depth, confidence, prob volume.
// All 48 logits live in VGPRs (fully unrolled); confidence uses a select-
// reduction to avoid scratch from a dynamic register index.
// ---------------------------------------------------------------------------
__global__ void __launch_bounds__(256) k_softmax(
    const float* __restrict__ cost,
    const float* __restrict__ depth_values,
    float* __restrict__ out_depth,
    float* __restrict__ out_conf,
    float* __restrict__ out_prob) {
  const int tid = blockIdx.x * blockDim.x + threadIdx.x;
  const int npix = B_ * HW_;
  if (tid >= npix) return;
  const int b = tid / HW_;
  const int hw = tid % HW_;
  const float* cb = cost + b * D_ * HW_ + hw;

  float e[D_];
  float mx = -3.402823466e38f;
#pragma unroll
  for (int dd = 0; dd < D_; ++dd) {
    e[dd] = cb[dd * HW_];
    mx = fmaxf(mx, e[dd]);
  }
  float sum = 0.f;
#pragma unroll
  for (int dd = 0; dd < D_; ++dd) {
    float ex = __expf(e[dd] - mx);
    e[dd] = ex;
    sum += ex;
  }
  const float inv = 1.0f / sum;
  float depv = 0.f, di = 0.f;
  float* pb = out_prob + b * D_ * HW_ + hw;
#pragma unroll
  for (int dd = 0; dd < D_; ++dd) {
    float pr = e[dd] * inv;
    pb[dd * HW_] = pr;
    depv += pr * depth_values[b * D_ + dd];
    di += pr * (float)dd;
  }
  int idx = (int)di;
  idx = min(max(idx, 0), D_ - 1);
  float conf = 0.f;
#pragma unroll
  for (int dd = 0; dd < D_; ++dd) {
    float pr = e[dd] * inv;
    conf += (dd == idx || dd == idx + 1) ? pr : 0.f;
  }
  out_depth[tid] = depv;
  out_conf[tid] = conf;
}

// ---------------------------------------------------------------------------
extern "C" void kernel_launch(void* const* d_in, const int* in_sizes, int n_in,
                              void* d_out, int out_size, void* d_ws, size_t ws_size,
                              hipStream_t stream) {
  (void)in_sizes; (void)n_in; (void)out_size; (void)ws_size;
  const float* ref_feature = (const float*)d_in[0];
  const float* src_features = (const float*)d_in[1];
  const float* ref_proj = (const float*)d_in[2];
  const float* src_projs = (const float*)d_in[3];
  const float* depth_values = (const float*)d_in[4];
  const float* w_reg = (const float*)d_in[5];
  const float* b_reg = (const float*)d_in[6];

  float* prm = (float*)d_ws;                      // 48 floats of proj params
  float* cost = (float*)((char*)d_ws + 1024);     // B*D*H*W cost volume

  float* out_depth = (float*)d_out;
  float* out_conf = out_depth + B_ * HW_;
  float* out_prob = out_conf + B_ * HW_;

  k_proj<<<1, 32, 0, stream>>>(ref_proj, src_projs, prm);

  const int points = B_ * D_ * HW_;               // 1,179,648
  const int blocks = points / (16 * 8);           // 16 pts/wave, 8 waves/block
  k_cost<<<blocks, 256, 0, stream>>>(ref_feature, src_features, depth_values,
                                     w_reg, b_reg, prm, cost);

  k_softmax<<<(B_ * HW_ + 255) / 256, 256, 0, stream>>>(
      cost, depth_values, out_depth, out_conf, out_prob);
}